// RelativeCoordPredictor_27032524161649
// MI455X (gfx1250) — compile-verified
//
#include <hip/hip_runtime.h>
#include <hip/hip_bf16.h>
#include <math.h>

// Problem constants (from reference: x = [32, 512, 32, 32] f32)
#define N_   32
#define C_   512
#define H_   32
#define W_   32
#define HW_  (H_ * W_)      // 1024
#define SIZE_ ((float)H_)   // 32.0f
#define PI_  3.14159265358979323846f

#define CSPLIT_ 4                       // channel-dim split for the reduction
#define CCHUNK_ (C_ / CSPLIT_)          // 128 channels per chunk

typedef float v2f __attribute__((ext_vector_type(2)));
typedef float v4f __attribute__((ext_vector_type(4)));
typedef float v8f __attribute__((ext_vector_type(8)));

// ---------------------------------------------------------------------------
// Kernel 1: partial channel sums with float4 (global_load_b128) streaming.
//   partial[chunk][n*HW + hw] = sum_{c in chunk} x[n,c,hw]
//   Grid: 32 pos-blocks x 4 chunks = 128 blocks, 256 threads each.
//   Each lane loads 16 B/iter -> wave streams 512 B per load instruction.
// ---------------------------------------------------------------------------
__global__ void k_mask_partial(const float* __restrict__ x,
                               float* __restrict__ partial) {
    const int POSBLK = (N_ * HW_ / 4) / 256;     // 32
    int chunk = blockIdx.x / POSBLK;             // 0..3
    int pos   = (blockIdx.x % POSBLK) * 256 + threadIdx.x;  // float4 position
    int n   = pos / (HW_ / 4);
    int hw4 = pos - n * (HW_ / 4);

    const v4f* p = (const v4f*)(x + (size_t)n * C_ * HW_
                                  + (size_t)chunk * CCHUNK_ * HW_) + hw4;
    v4f s = {0.f, 0.f, 0.f, 0.f};
#pragma unroll 4
    for (int c = 0; c < CCHUNK_; ++c) {
        s += p[(size_t)c * (HW_ / 4)];           // coalesced b128 across wave
    }
    ((v4f*)partial)[(size_t)chunk * (N_ * HW_ / 4) + pos] = s;
}

// ---------------------------------------------------------------------------
// Kernel 2: per-sample stats. One block (1024 threads = 32 waves) per n.
//   mask = sum of 4 partials; thr = mean(mask); bm = mask>thr;
//   mm = bm*mask/C (masked_mean identity); argmax -> anchor;
//   per-pixel dist/angle -> relative_coord_total.
// ---------------------------------------------------------------------------
__global__ void k_stats(const float* __restrict__ partial,
                        float* __restrict__ mm_ws,
                        float* __restrict__ rct,      // [N, HW, 2]
                        float* __restrict__ anchor,   // [N, 2]
                        float* __restrict__ idx_out)  // [N] (as float)
{
    __shared__ float redf[HW_];
    __shared__ int   redi[HW_];

    int n = blockIdx.x;
    int t = threadIdx.x;            // == hw, 0..1023

    float m = 0.f;
#pragma unroll
    for (int p = 0; p < CSPLIT_; ++p) {
        m += partial[(size_t)p * (N_ * HW_) + n * HW_ + t];
    }

    // --- sum reduce for threshold ---
    redf[t] = m;
    __syncthreads();
    for (int s = HW_ / 2; s > 0; s >>= 1) {
        if (t < s) redf[t] += redf[t + s];
        __syncthreads();
    }
    float thr = redf[0] * (1.0f / (float)HW_);
    __syncthreads();

    float bm = (m > thr) ? 1.0f : 0.0f;
    float mm = bm * m * (1.0f / (float)C_);   // masked_mean (algebraic identity)
    mm_ws[n * HW_ + t] = mm;

    // --- argmax reduce (first occurrence of max, matching jnp.argmax) ---
    redf[t] = mm;
    redi[t] = t;
    __syncthreads();
    for (int s = HW_ / 2; s > 0; s >>= 1) {
        if (t < s) {
            float v2 = redf[t + s]; int i2 = redi[t + s];
            if (v2 > redf[t] || (v2 == redf[t] && i2 < redi[t])) {
                redf[t] = v2; redi[t] = i2;
            }
        }
        __syncthreads();
    }
    int amax = redi[0];
    float ai = (float)(amax / W_);
    float aj = (float)(amax % W_);
    if (t == 0) {
        anchor[n * 2 + 0] = ai;
        anchor[n * 2 + 1] = aj;
        idx_out[n] = (float)amax;
    }

    // --- relative coord: dist & angle, masked ---
    float ci = ((float)(t / W_) - ai) * (1.0f / SIZE_);
    float cj = ((float)(t % W_) - aj) * (1.0f / SIZE_);
    float dist = sqrtf(ci * ci + cj * cj) * bm;
    float ang  = ((atan2f(cj, ci) * (1.0f / PI_)) + 1.0f) * 0.5f * bm;
    size_t o = ((size_t)n * HW_ + t) * 2;
    rct[o + 0] = dist;
    rct[o + 1] = ang;
}

// ---------------------------------------------------------------------------
// Kernel 3: position_weight[n] = mm[n] (outer) mm[n]  via V_WMMA_F32_16X16X4_F32
//   Rank-1 tile: A has mm[i0..i0+15] in K=0 column, B has mm[j0..j0+15] in
//   K=0 row; everything else zero. All lanes load (lane&15 aliases the same
//   64B line), then v_cndmask selects -> NO exec-mask branching, EXEC stays
//   all-1s through every WMMA.
//   D layout: lane<16 -> (M=r, N=lane); lane>=16 -> (M=r+8, N=lane-16).
//   Grid: N*64 blocks (one i-tile each), 8 waves/block, 8 j-tiles per wave.
// ---------------------------------------------------------------------------
__global__ void k_outer_wmma(const float* __restrict__ mm, float* __restrict__ pw) {
    int blk  = blockIdx.x;              // 0 .. N_*64-1
    int n    = blk >> 6;
    int ti   = blk & 63;                // i-tile index
    int wave = threadIdx.x >> 5;        // 0..7
    int lane = threadIdx.x & 31;
    int laneN = lane & 15;
    bool lo  = lane < 16;

    int i0 = ti * 16;
    const float* mrow = mm + n * HW_;

    float av = mrow[i0 + laneN];        // unconditional, both halves same line
    v2f a;
    a.x = lo ? av : 0.0f;               // K=0 column of A (v_cndmask, no branch)
    a.y = 0.0f;

    int   rowoff = lo ? 0 : 8;          // lanes 16-31 cover rows M=8..15
    float* obase = pw + ((size_t)n * HW_ + (i0 + rowoff)) * HW_;

#pragma unroll
    for (int jt = 0; jt < 8; ++jt) {
        int j0 = (wave * 8 + jt) * 16;
        float bv = mrow[j0 + laneN];    // unconditional load
        v2f b;
        b.x = lo ? bv : 0.0f;           // K=0 row of B
        b.y = 0.0f;
        v8f c = {};
        // 8 args: (neg_a, A, neg_b, B, c_mod, C, reuse_a, reuse_b)
        c = __builtin_amdgcn_wmma_f32_16x16x4_f32(
                false, a, false, b, (short)0, c, false, false);
        float* o = obase + j0 + laneN;
#pragma unroll
        for (int r = 0; r < 8; ++r) {
            o[(size_t)r * HW_] = c[r];  // row M = i0+rowoff+r, col N = j0+laneN
        }
    }
}

// ---------------------------------------------------------------------------
// Launch. Output layout (flat, return order):
//   rct [N*HW*2] | anchor [N*2] | position_weight [N*HW*HW] | index [N]
// Workspace: partial [4*N*HW] | mm [N*HW]   (640 KiB)
// ---------------------------------------------------------------------------
extern "C" void kernel_launch(void* const* d_in, const int* in_sizes, int n_in,
                              void* d_out, int out_size, void* d_ws, size_t ws_size,
                              hipStream_t stream) {
    const float* x = (const float*)d_in[0];
    float* out = (float*)d_out;

    float* partial = (float*)d_ws;
    float* mm      = partial + (size_t)CSPLIT_ * N_ * HW_;

    float* rct    = out;
    float* anchor = rct + (size_t)N_ * HW_ * 2;
    float* pw     = anchor + (size_t)N_ * 2;
    float* idxo   = pw + (size_t)N_ * HW_ * HW_;

    k_mask_partial<<<(N_ * HW_ / 4 / 256) * CSPLIT_, 256, 0, stream>>>(x, partial);
    k_stats<<<N_, HW_, 0, stream>>>(partial, mm, rct, anchor, idxo);
    k_outer_wmma<<<N_ * 64, 256, 0, stream>>>(mm, pw);
}